// Dynamic_spiral_pool_21878563406305
// MI455X (gfx1250) — compile-verified
//
#include <hip/hip_runtime.h>
#include <hip/hip_bf16.h>
#include <stdint.h>

// Problem constants (reference: B=4, N=50000, C=64, K=16, G=4)
#define Bb 4
#define Nn 50000
#define Cc 64
#define Kk 16
#define Gg 4
#define NSLOT 256            // slot-hashed partial accumulators per (b,g,stat)
#define EPSV 1e-5f

typedef unsigned int u32;
typedef unsigned int v4u __attribute__((ext_vector_type(4)));
typedef int          v8i __attribute__((ext_vector_type(8)));
typedef int          v4i __attribute__((ext_vector_type(4)));

// ---------------------------------------------------------------------------
// Kernel 0: zero the partial-sum workspace (32 combos x NSLOT slots)
// ---------------------------------------------------------------------------
__global__ __launch_bounds__(256) void zero_partials(float* __restrict__ p) {
    int i = blockIdx.x * 256 + threadIdx.x;   // grid sized to exactly 32*NSLOT
    p[i] = 0.0f;
}

// ---------------------------------------------------------------------------
// Kernel 1: TDM-gather + learned pooling.  One wave32 per point n, looping
// over the 4 batches with double-buffered tensor_load_to_lds (gather mode).
// Writes un-normalized y to d_out and accumulates per-(b,group) sum/sumsq.
// ---------------------------------------------------------------------------
__global__ __launch_bounds__(256) void spiral_pool_kernel(
    const float* __restrict__ x,          // [B,N,C]
    const long long* __restrict__ indices,// [N,K] int64
    const float* __restrict__ ro_w,       // [1,C]
    const float* __restrict__ ro_b,       // [1]
    float* __restrict__ yout,             // [B,N,C]
    float* __restrict__ partial)          // [32][NSLOT]
{
    // 8 waves * 2 buffers * (16 rows * 64 ch) floats = 64 KB LDS
    __shared__ float smem[8 * 2 * Kk * Cc];

    const int tid  = threadIdx.x;
    const int wave = tid >> 5;
    const int lane = tid & 31;
    const int n    = blockIdx.x * 8 + wave;      // grid = N/8 -> always < N

    const float w0 = ro_w[2 * lane];
    const float w1 = ro_w[2 * lane + 1];
    const float rb = ro_b[0];

    // ---- load K=16 int64 indices for this point, pack to 16x u16 ----------
    u32 myidx = 0;
    if (lane < Kk)
        myidx = (u32)(unsigned long long)indices[(size_t)n * Kk + lane];
    u32 p0, p1, p2, p3, p4, p5, p6, p7;
    {
        u32 q[8];
#pragma unroll
        for (int j = 0; j < 8; ++j) {
            u32 lo = (u32)__builtin_amdgcn_readlane((int)myidx, 2 * j);
            u32 hi = (u32)__builtin_amdgcn_readlane((int)myidx, 2 * j + 1);
            q[j] = (lo & 0xFFFFu) | (hi << 16);
        }
        p0 = q[0]; p1 = q[1]; p2 = q[2]; p3 = q[3];
        p4 = q[4]; p5 = q[5]; p6 = q[6]; p7 = q[7];
    }

    // LDS byte offset of smem (low 32 bits of a generic LDS pointer)
    const u32 ldsBase = (u32)(uintptr_t)&smem[0];

    // ---- TDM gather-mode descriptor issue ---------------------------------
    // D# group0: count=1, gather_mode=1 (16-bit indices), lds_addr,
    //            global_addr (57b), type=2.
    // D# group1: data_size=4B, tensor_dim0=64, tensor_dim1=N, tile_dim0=64,
    //            tile_dim1=16 (#indices), tensor_dim0_stride=64.
    auto issue = [&](int b, int buf) {
        unsigned long long ga =
            (unsigned long long)(uintptr_t)(x + (size_t)b * (size_t)Nn * Cc);
        v4u g0;
        g0.x = 0x80000001u;                                   // gather, count=1
        g0.y = ldsBase + (u32)((wave * 2 + buf) * (Kk * Cc * 4));
        g0.z = (u32)ga;
        g0.w = ((u32)(ga >> 32) & 0x01FFFFFFu) | 0x80000000u; // type=2
        v8i g1 = { (int)0x00020000,        // data_size = 2 (4 bytes)
                   (int)(64u << 16),       // tensor_dim0 = 64  (bits 79:48)
                   (int)(50000u << 16),    // tensor_dim1 = N   (bits 111:80)
                   (int)(64u << 16),       // tile_dim0  = 64   (bits 127:112)
                   (int)16,                // tile_dim1  = 16 valid indices
                   (int)64,                // tensor_dim0_stride = 64 elems
                   0, 0 };
        v4i g2 = { (int)p0, (int)p1, (int)p2, (int)p3 };      // rows 0..7
        v4i g3 = { (int)p4, (int)p5, (int)p6, (int)p7 };      // rows 8..15
        v8i g4 = { 0, 0, 0, 0, 0, 0, 0, 0 };                  // VADDR4: unused
        __builtin_amdgcn_tensor_load_to_lds(g0, g1, g2, g3, g4, 0);
    };

    // ---- per-batch processing of one gathered tile -------------------------
    auto process = [&](int b, int buf) {
        const float* tile = &smem[(wave * 2 + buf) * (Kk * Cc)];
        float gA[Kk], gB[Kk];
        float s0 = 0.0f, s1 = 0.0f;
#pragma unroll
        for (int k = 0; k < Kk; ++k) {
            float2 v = *(const float2*)(tile + k * Cc + 2 * lane);
            gA[k] = v.x; gB[k] = v.y;
            s0 += v.x;   s1 += v.y;
        }
        // s = |mean . ro_w + ro_b|, clamp, split into ib/frac
        float dot = s0 * (1.0f / 16.0f) * w0 + s1 * (1.0f / 16.0f) * w1;
#pragma unroll
        for (int m = 1; m < 32; m <<= 1) dot += __shfl_xor(dot, m, 32);
        float s   = fabsf(dot + rb);
        s         = fminf(s * 16.0f, 15.0f);
        float fib = floorf(s);
        int   ib  = (int)fib;
        float fr  = s - fib;
        // y = pool[ib] + frac * g[ib+1]   (== lerp of cumsum)
        float a0 = 0.0f, a1 = 0.0f;
#pragma unroll
        for (int k = 0; k < Kk; ++k) {
            float w = (k <= ib) ? 1.0f : ((k == ib + 1) ? fr : 0.0f);
            a0 = fmaf(gA[k], w, a0);
            a1 = fmaf(gB[k], w, a1);
        }
        float2 o; o.x = a0; o.y = a1;
        *(float2*)(yout + ((size_t)b * Nn + n) * Cc + 2 * lane) = o;

        // group stats: channels 2*lane,2*lane+1 -> group = lane/8
        float ls = a0 + a1;
        float lq = a0 * a0 + a1 * a1;
        ls += __shfl_xor(ls, 1, 32); ls += __shfl_xor(ls, 2, 32); ls += __shfl_xor(ls, 4, 32);
        lq += __shfl_xor(lq, 1, 32); lq += __shfl_xor(lq, 2, 32); lq += __shfl_xor(lq, 4, 32);
        if ((lane & 7) == 0) {
            int tg   = b * Gg + (lane >> 3);
            int slot = n & (NSLOT - 1);
            atomicAdd(partial + (size_t)(2 * tg + 0) * NSLOT + slot, ls);
            atomicAdd(partial + (size_t)(2 * tg + 1) * NSLOT + slot, lq);
        }
    };

    // ---- double-buffered pipeline over the 4 batches -----------------------
    issue(0, 0);
    issue(1, 1);
    __builtin_amdgcn_s_wait_tensorcnt(1);   // tile(b=0) resident
    process(0, 0);
    issue(2, 0);
    __builtin_amdgcn_s_wait_tensorcnt(1);   // tile(b=1) resident
    process(1, 1);
    issue(3, 1);
    __builtin_amdgcn_s_wait_tensorcnt(1);   // tile(b=2) resident
    process(2, 0);
    __builtin_amdgcn_s_wait_tensorcnt(0);   // tile(b=3) resident
    process(3, 1);
}

// ---------------------------------------------------------------------------
// Kernel 2: fold NSLOT partials -> mean / inv-std per (b,group)
// ---------------------------------------------------------------------------
__global__ __launch_bounds__(64) void reduce_stats(
    const float* __restrict__ partial, float* __restrict__ mstd)
{
    __shared__ float red[32];
    int t = threadIdx.x;
    if (t < 32) {
        float s = 0.0f;
        for (int i = 0; i < NSLOT; ++i) s += partial[(size_t)t * NSLOT + i];
        red[t] = s;
    }
    __syncthreads();
    if (t < 16) {
        const float invCnt = 1.0f / ((float)Nn * (Cc / Gg));   // 1/800000
        float S    = red[2 * t];
        float Q    = red[2 * t + 1];
        float mean = S * invCnt;
        float var  = fmaxf(Q * invCnt - mean * mean, 0.0f);
        mstd[t]      = mean;
        mstd[16 + t] = rsqrtf(var + EPSV);
    }
}

// ---------------------------------------------------------------------------
// Kernel 3: in-place GroupNorm apply, float4-vectorized
// ---------------------------------------------------------------------------
__global__ __launch_bounds__(256) void gn_apply(
    float* __restrict__ y, const float* __restrict__ mstd,
    const float* __restrict__ gamma, const float* __restrict__ beta)
{
    __shared__ float sm[32];
    if (threadIdx.x < 32) sm[threadIdx.x] = mstd[threadIdx.x];
    __syncthreads();
    size_t i = ((size_t)blockIdx.x * 256 + threadIdx.x) * 4;
    int c = (int)(i & (Cc - 1));
    int b = (int)((i >> 6) / Nn);
    int t = b * Gg + (c >> 4);            // 4 consecutive c stay in one group
    float mean = sm[t], istd = sm[16 + t];
    float4 v  = *(float4*)(y + i);
    float4 gm = *(const float4*)(gamma + c);
    float4 bt = *(const float4*)(beta + c);
    v.x = (v.x - mean) * istd * gm.x + bt.x;
    v.y = (v.y - mean) * istd * gm.y + bt.y;
    v.z = (v.z - mean) * istd * gm.z + bt.z;
    v.w = (v.w - mean) * istd * gm.w + bt.w;
    *(float4*)(y + i) = v;
}

// ---------------------------------------------------------------------------
extern "C" void kernel_launch(void* const* d_in, const int* in_sizes, int n_in,
                              void* d_out, int out_size, void* d_ws, size_t ws_size,
                              hipStream_t stream) {
    const float*     x     = (const float*)d_in[0];
    const long long* idx   = (const long long*)d_in[1];   // int64 indices
    const float*     ro_w  = (const float*)d_in[2];
    const float*     ro_b  = (const float*)d_in[3];
    const float*     gamma = (const float*)d_in[4];
    const float*     beta  = (const float*)d_in[5];
    float*           out   = (float*)d_out;

    float* partial = (float*)d_ws;                 // 32 * NSLOT floats
    float* mstd    = partial + 32 * NSLOT;         // 16 means + 16 inv-stds

    zero_partials<<<(32 * NSLOT) / 256, 256, 0, stream>>>(partial);
    spiral_pool_kernel<<<Nn / 8, 256, 0, stream>>>(x, idx, ro_w, ro_b, out, partial);
    reduce_stats<<<1, 64, 0, stream>>>(partial, mstd);
    gn_apply<<<(Bb * (size_t)Nn * Cc) / 4 / 256, 256, 0, stream>>>(out, mstd, gamma, beta);
}